// CorrBlock_28741921145254
// MI455X (gfx1250) — compile-verified
//
#include <hip/hip_runtime.h>
#include <hip/hip_bf16.h>

typedef __attribute__((ext_vector_type(2))) float v2f;
typedef __attribute__((ext_vector_type(8))) float v8f;

#define EPS_    1e-5f
#define N_      4
#define C_      256
#define H_      192
#define W_      192
#define R_      7
#define S_      49
#define HW_     (H_ * W_)
#define CHW_    (C_ * HW_)
#define QW_     8            // query block width  (x)
#define QH_     2            // query block height (y)
#define KW_     14           // key halo width  = QW_ + R_ - 1
#define KH_     8            // key halo height = QH_ + R_ - 1
#define NK_     112          // KW_*KH_ = exactly 7 WMMA B-tiles of 16
#define OUT_PLANE (N_ * S_ * HW_)   // 7,225,344 floats per output tensor
#define WAVE_LDS  (16 * NK_ + NK_)  // G tile + key inv-norms, per wave

__global__ __launch_bounds__(128) void corr_wmma_kernel(
    const float* __restrict__ img,
    const float* __restrict__ map,
    float* __restrict__ out)
{
    const int tid  = threadIdx.x;
    const int wv   = tid >> 5;     // wave id within block (4 waves)
    const int lane = tid & 31;
    const int hh   = lane >> 4;    // K-subgroup: lanes 0-15 -> K{0,1}, 16-31 -> K{2,3}
    const int m16  = lane & 15;    // M row (A/queries) and N col (B/keys) index

    const int n  = blockIdx.z;
    const int x0 = blockIdx.x * QW_;
    const int y0 = (blockIdx.y * 4 + wv) * QH_;

    // this lane's query pixel (M row m16)
    const int qy = m16 >> 3, qx = m16 & 7;
    const int yq = y0 + qy,  xq = x0 + qx;

    const float* aP = img + (size_t)n * CHW_ + (size_t)(2 * hh) * HW_ + yq * W_ + xq;
    const float* bP = map + (size_t)n * CHW_ + (size_t)(2 * hh) * HW_;

    // key pixel offsets for the 7 B-tiles (edge clamp: only low side can underflow)
    int koff[7];
#pragma unroll
    for (int t = 0; t < 7; ++t) {
        int kk = 16 * t + m16;             // 0..111, exactly covers 8x14 halo
        int ry = kk / KW_, rx = kk % KW_;
        int ky = y0 - R_ + ry; ky = ky < 0 ? 0 : ky;
        int kx = x0 - R_ + rx; kx = kx < 0 ? 0 : kx;
        koff[t] = ky * W_ + kx;
    }

    v8f  acc[7] = {};                      // G[16x112] fp32 accumulators
    float asum = 0.f;                      // sum img^2 (this lane's channel half)
    float ksum[7] = {0.f, 0.f, 0.f, 0.f, 0.f, 0.f, 0.f};

#pragma unroll 4
    for (int c0 = 0; c0 < C_; c0 += 4) {
        const int co = c0 * HW_;
        float a0 = aP[co];
        float a1 = aP[co + HW_];
        asum += a0 * a0 + a1 * a1;
        v2f A; A.x = a0; A.y = a1;
#pragma unroll
        for (int t = 0; t < 7; ++t) {
            float b0 = bP[co + koff[t]];
            float b1 = bP[co + HW_ + koff[t]];
            ksum[t] += b0 * b0 + b1 * b1;
            v2f B; B.x = b0; B.y = b1;
            // D = A(16x4, f32) x B(4x16, f32) + C  -> v_wmma_f32_16x16x4_f32
            acc[t] = __builtin_amdgcn_wmma_f32_16x16x4_f32(
                false, A, false, B, (short)0, acc[t], false, false);
        }
    }

    // Combine the two channel-halves (lane ^ 16) of the sum-of-squares.
    asum += __int_as_float(
        __builtin_amdgcn_ds_swizzle(__float_as_int(asum), 0x401F)); // xor 0x10
    const float qinv = 1.f / (sqrtf(asum) + EPS_);

    float kinv[7];
#pragma unroll
    for (int t = 0; t < 7; ++t) {
        float s2 = ksum[t] + __int_as_float(
            __builtin_amdgcn_ds_swizzle(__float_as_int(ksum[t]), 0x401F));
        kinv[t] = 1.f / (sqrtf(s2) + EPS_);
    }

    // Spill G and key inv-norms to this wave's LDS region for banded extraction.
    __shared__ float lds[4 * WAVE_LDS];
    float* G     = lds + wv * WAVE_LDS;
    float* kinvA = G + 16 * NK_;
#pragma unroll
    for (int t = 0; t < 7; ++t) {
#pragma unroll
        for (int j = 0; j < 8; ++j) {
            int m = j + 8 * hh;            // D layout: comp j, lane half hh -> row
            G[m * NK_ + 16 * t + m16] = acc[t][j];
        }
        if (hh == 0) kinvA[16 * t + m16] = kinv[t];
    }
    __syncthreads();

    // Extraction: out[s, q] from the band G[q, (qy+i)*14 + (qx+j)].
    float* outCos  = out;
    float* outDiss = out + (size_t)OUT_PLANE;
    const int pixOff = yq * W_ + xq;
    const int q = m16;
    for (int s = hh; s < S_; s += 2) {
        int i  = s / 7, j = s - 7 * i;
        int kk = (qy + i) * KW_ + (qx + j);
        float D  = G[q * NK_ + kk];
        float d1 = D * qinv;                     // dot(map_raw, img_n)
        float cv = 1.0f - d1 * kinvA[kk];        // 1 - dot(s_map, s_img)
        int o = (n * S_ + s) * HW_ + pixOff;
        __builtin_nontemporal_store(cv, outCos + o);
        __builtin_nontemporal_store(d1, outDiss + o);
    }
}

extern "C" void kernel_launch(void* const* d_in, const int* in_sizes, int n_in,
                              void* d_out, int out_size, void* d_ws, size_t ws_size,
                              hipStream_t stream) {
    const float* img = (const float*)d_in[0];
    const float* map = (const float*)d_in[1];
    // d_in[2] is radius == 7 (baked into the tiling as R_)
    (void)in_sizes; (void)n_in; (void)out_size; (void)d_ws; (void)ws_size;

    dim3 grid(W_ / QW_, H_ / (QH_ * 4), N_);   // 24 x 24 x 4 blocks
    corr_wmma_kernel<<<grid, dim3(128), 0, stream>>>(img, map, (float*)d_out);
}